// ParticleFilterNetwork_2911987827277
// MI455X (gfx1250) — compile-verified
//
#include <hip/hip_runtime.h>
#include <cstdint>

// ---------------------------------------------------------------------------
// Particle filter step for MI455X (gfx1250, wave32).
// Bandwidth-bound (~0.9 GB HBM traffic @ 23.3 TB/s ≈ 40us); matrix work is
// free, so the D=4 dynamics matmul is mapped onto V_WMMA_F32_16X16X4_F32.
// Streaming inputs are loaded non-temporal so the 192MB L2 keeps the
// intermediate spred/CDF (which the resampling gather re-reads) resident.
// Resampling stages each 64KB CDF row in LDS via the CDNA5 async
// global->LDS path (ASYNCcnt) and binary-searches in LDS.
// ---------------------------------------------------------------------------

typedef float v2f __attribute__((ext_vector_type(2)));
typedef float v4f __attribute__((ext_vector_type(4)));
typedef float v8f __attribute__((ext_vector_type(8)));

#define NROWS 512
#define MPART 16384
#define SDIM 4
#define OBSD 16
#define CTRLD 4
#define BLOCKS_PER_ROW 8
#define WAVES 8
#define PER_WAVE (MPART / BLOCKS_PER_ROW / WAVES) // 256 particles per wave

// ---------------------------------------------------------------------------
// Kernel 1: per-row constants  b = controls@B ,  obsfeat = observations@W
// ---------------------------------------------------------------------------
__global__ void k_rowconst(const float* __restrict__ obs,
                           const float* __restrict__ ctrl,
                           const float* __restrict__ Bm,
                           const float* __restrict__ Wm,
                           float* __restrict__ rowc) {
  int n = blockIdx.x * blockDim.x + threadIdx.x;
  if (n >= NROWS) return;
#pragma unroll
  for (int e = 0; e < SDIM; ++e) {
    float b = 0.f;
#pragma unroll
    for (int d = 0; d < CTRLD; ++d) b += ctrl[n * CTRLD + d] * Bm[d * SDIM + e];
    float of = 0.f;
#pragma unroll
    for (int d = 0; d < OBSD; ++d) of += obs[n * OBSD + d] * Wm[d * SDIM + e];
    rowc[n * 8 + e] = b;
    rowc[n * 8 + 4 + e] = of;
  }
}

// ---------------------------------------------------------------------------
// Kernel 2: main pass. WMMA dynamics + noise + log-lik + online logsumexp /
// weighted-state partials per block. 32 particles per wave-iteration.
// ---------------------------------------------------------------------------
__global__ __launch_bounds__(256) void k_main(
    const float* __restrict__ sprev, const float* __restrict__ lwprev,
    const float* __restrict__ noise, const float* __restrict__ Adyn,
    const float* __restrict__ rowc, float* __restrict__ spred,
    float* __restrict__ lwout, float* __restrict__ partials) {
  const int n = blockIdx.x / BLOCKS_PER_ROW;
  const int blk = blockIdx.x % BLOCKS_PER_ROW;
  const int lane = threadIdx.x & 31;
  const int wave = threadIdx.x >> 5;
  const int mrow = lane & 15;
  const int half = lane >> 4;

  // WMMA A-operand: rows = output state dims (4 valid of 16), K = input dim.
  // ISA layout (16x4 f32 A): VGPR0 = K{0|2}, VGPR1 = K{1|3} by half-wave.
  // Amat[m][k] = Adyn[k][m]  (einsum 'nmd,de->nme' => out_e = sum_d s_d*A[d][e])
  v2f amat;
  {
    const bool valid = (mrow < SDIM);
    amat[0] = valid ? Adyn[(half * 2 + 0) * SDIM + mrow] : 0.0f;
    amat[1] = valid ? Adyn[(half * 2 + 1) * SDIM + mrow] : 0.0f;
  }

  const float b0 = rowc[n * 8 + 0], b1 = rowc[n * 8 + 1];
  const float b2 = rowc[n * 8 + 2], b3 = rowc[n * 8 + 3];
  const float o0 = rowc[n * 8 + 4], o1 = rowc[n * 8 + 5];
  const float o2 = rowc[n * 8 + 6], o3 = rowc[n * 8 + 7];

  const size_t rowbase = (size_t)n * MPART;
  const int mstart = blk * (MPART / BLOCKS_PER_ROW) + wave * PER_WAVE;

  float mx = -3.0e38f, se = 0.f, w0 = 0.f, w1 = 0.f, w2 = 0.f, w3 = 0.f;

#pragma unroll 2
  for (int it = 0; it < PER_WAVE; it += 32) {
    const int m0 = mstart + it;
    const float* tile = sprev + (rowbase + m0) * SDIM;
    // prefetch next tile (speculative; OOB prefetch is silently dropped)
    __builtin_prefetch(tile + 32 * SDIM, 0, 0);
    __builtin_prefetch(noise + (rowbase + m0 + 32) * SDIM, 0, 0);

    // B-operands (K x N = 4 x 16): lane p holds s[p][0..1], lane p+16 holds
    // s[p][2..3]  -> fully coalesced b64 loads covering 16 particles each.
    // Streamed data is read exactly once -> non-temporal.
    const v2f bA = __builtin_nontemporal_load(
        (const v2f*)(tile + mrow * SDIM + half * 2));
    const v2f bB = __builtin_nontemporal_load(
        (const v2f*)(tile + (16 + mrow) * SDIM + half * 2));

    v8f c0 = {};
    c0 = __builtin_amdgcn_wmma_f32_16x16x4_f32(false, amat, false, bA,
                                               (short)0, c0, false, false);
    v8f c1 = {};
    c1 = __builtin_amdgcn_wmma_f32_16x16x4_f32(false, amat, false, bB,
                                               (short)0, c1, false, false);

    // lane q < 16 : particle m0+q    -> c0[d] at own lane
    // lane q >= 16: particle m0+q    -> c1[d] lives at lane q-16, shuffle up
    float s0, s1, s2, s3;
    {
      float t0 = __shfl(c1[0], mrow, 32); s0 = half ? t0 : c0[0];
      float t1 = __shfl(c1[1], mrow, 32); s1 = half ? t1 : c0[1];
      float t2 = __shfl(c1[2], mrow, 32); s2 = half ? t2 : c0[2];
      float t3 = __shfl(c1[3], mrow, 32); s3 = half ? t3 : c0[3];
    }

    const v4f nz = __builtin_nontemporal_load(
        (const v4f*)(noise + (rowbase + m0 + lane) * SDIM));
    s0 += b0 + nz[0]; s1 += b1 + nz[1]; s2 += b2 + nz[2]; s3 += b3 + nz[3];

    // spred / lw are re-read by later kernels -> keep regular (L2-resident)
    v4f st; st[0] = s0; st[1] = s1; st[2] = s2; st[3] = s3;
    *(v4f*)(spred + (rowbase + m0 + lane) * SDIM) = st;

    const float d0 = s0 - o0, d1 = s1 - o1, d2 = s2 - o2, d3 = s3 - o3;
    const float ll = -0.5f * (d0 * d0 + d1 * d1 + d2 * d2 + d3 * d3);
    const float lw =
        __builtin_nontemporal_load(lwprev + (rowbase + m0 + lane)) + ll;
    lwout[rowbase + m0 + lane] = lw;

    // branchless online logsumexp + weighted-state accumulation
    const float m = fmaxf(mx, lw);
    const float sc = __expf(mx - m);
    const float e = __expf(lw - m);
    se = se * sc + e;
    w0 = w0 * sc + e * s0; w1 = w1 * sc + e * s1;
    w2 = w2 * sc + e * s2; w3 = w3 * sc + e * s3;
    mx = m;
  }

  // wave-level merge (shfl_xor tree over 32 lanes)
#pragma unroll
  for (int off = 16; off > 0; off >>= 1) {
    float m2 = __shfl_xor(mx, off, 32);
    float se2 = __shfl_xor(se, off, 32);
    float a0 = __shfl_xor(w0, off, 32);
    float a1 = __shfl_xor(w1, off, 32);
    float a2 = __shfl_xor(w2, off, 32);
    float a3 = __shfl_xor(w3, off, 32);
    float m = fmaxf(mx, m2);
    float sc1 = __expf(mx - m), sc2 = __expf(m2 - m);
    se = se * sc1 + se2 * sc2;
    w0 = w0 * sc1 + a0 * sc2; w1 = w1 * sc1 + a1 * sc2;
    w2 = w2 * sc1 + a2 * sc2; w3 = w3 * sc1 + a3 * sc2;
    mx = m;
  }

  __shared__ float sm[WAVES][6];
  if (lane == 0) {
    sm[wave][0] = mx; sm[wave][1] = se;
    sm[wave][2] = w0; sm[wave][3] = w1; sm[wave][4] = w2; sm[wave][5] = w3;
  }
  __syncthreads();
  if (threadIdx.x == 0) {
    float Mx = -3.0e38f, S = 0.f, W0 = 0.f, W1 = 0.f, W2 = 0.f, W3 = 0.f;
#pragma unroll
    for (int v = 0; v < WAVES; ++v) {
      float m2 = sm[v][0], s2 = sm[v][1];
      float m = fmaxf(Mx, m2);
      float sc1 = __expf(Mx - m), sc2 = __expf(m2 - m);
      S = S * sc1 + s2 * sc2;
      W0 = W0 * sc1 + sm[v][2] * sc2; W1 = W1 * sc1 + sm[v][3] * sc2;
      W2 = W2 * sc1 + sm[v][4] * sc2; W3 = W3 * sc1 + sm[v][5] * sc2;
      Mx = m;
    }
    float* p = partials + ((size_t)n * BLOCKS_PER_ROW + blk) * 6;
    p[0] = Mx; p[1] = S; p[2] = W0; p[3] = W1; p[4] = W2; p[5] = W3;
  }
}

// ---------------------------------------------------------------------------
// Kernel 3: merge per-block partials -> lse[n] and best_states[n]
// ---------------------------------------------------------------------------
__global__ void k_finalize(const float* __restrict__ partials,
                           float* __restrict__ best,
                           float* __restrict__ lsews) {
  int n = blockIdx.x * blockDim.x + threadIdx.x;
  if (n >= NROWS) return;
  float Mx = -3.0e38f, S = 0.f, W[4] = {0.f, 0.f, 0.f, 0.f};
#pragma unroll
  for (int b = 0; b < BLOCKS_PER_ROW; ++b) {
    const float* p = partials + ((size_t)n * BLOCKS_PER_ROW + b) * 6;
    float m2 = p[0], s2 = p[1];
    float m = fmaxf(Mx, m2);
    float sc1 = __expf(Mx - m), sc2 = __expf(m2 - m);
    S = S * sc1 + s2 * sc2;
#pragma unroll
    for (int d = 0; d < SDIM; ++d) W[d] = W[d] * sc1 + p[2 + d] * sc2;
    Mx = m;
  }
  lsews[n] = Mx + __logf(S);
  const float inv = 1.0f / S; // sum exp(lw-lse) == 1, so best = W/S
#pragma unroll
  for (int d = 0; d < SDIM; ++d) best[n * SDIM + d] = W[d] * inv;
}

// ---------------------------------------------------------------------------
// Kernel 4: in-place per-row inclusive CDF of exp(lw - lse)
// ---------------------------------------------------------------------------
__global__ __launch_bounds__(256) void k_cdf(float* __restrict__ lw,
                                             const float* __restrict__ lsews) {
  const int n = blockIdx.x;
  const int t = threadIdx.x;
  const int PER = MPART / 256; // 64
  const float lse = lsews[n];
  float* row = lw + (size_t)n * MPART;
  const int base = t * PER;

  float tsum = 0.f;
  for (int i = 0; i < PER; ++i) tsum += __expf(row[base + i] - lse);

  __shared__ float sm[256];
  sm[t] = tsum;
  __syncthreads();
  for (int off = 1; off < 256; off <<= 1) {
    float add = (t >= off) ? sm[t - off] : 0.0f;
    __syncthreads();
    sm[t] += add;
    __syncthreads();
  }
  float run = (t == 0) ? 0.0f : sm[t - 1];
  for (int i = 0; i < PER; ++i) {
    run += __expf(row[base + i] - lse);
    row[base + i] = run; // overwrite lw with inclusive CDF
  }
}

// ---------------------------------------------------------------------------
// Kernel 5: multinomial resample. One block per row n: stage the 64KB CDF
// row in LDS via async global->LDS copies (ASYNCcnt), then 64 samples per
// thread with LDS binary search + L2 gather of spred. Outputs stored NT.
// ---------------------------------------------------------------------------
__device__ __forceinline__ float u01(uint32_t x) {
  x ^= x >> 17; x *= 0xed5ad4bbu;
  x ^= x >> 11; x *= 0xac4c1b51u;
  x ^= x >> 15; x *= 0x31848babu;
  x ^= x >> 14;
  return ((x >> 8) + 0.5f) * (1.0f / 16777216.0f);
}

__global__ __launch_bounds__(256) void k_resample(
    const float* __restrict__ spred, const float* __restrict__ cdf,
    float* __restrict__ out_states, float* __restrict__ out_lw) {
  const int n = blockIdx.x;
  const int t = threadIdx.x;
  __shared__ float scdf[MPART]; // 64 KB of the 320 KB/WGP LDS
  const float* row = cdf + ((size_t)n << 14);

  // Async bulk copy global->LDS: 16 x b128 chunks per thread, coalesced.
#pragma unroll
  for (int k = 0; k < MPART / (256 * 4); ++k) {
    const int c = k * 256 + t; // 16-byte chunk id
    uint32_t ldsoff = (uint32_t)(uintptr_t)(&scdf[c * 4]);
    const float* g = row + c * 4;
    asm volatile("global_load_async_to_lds_b128 %0, %1, off"
                 :
                 : "v"(ldsoff), "v"(g)
                 : "memory");
  }
  asm volatile("s_wait_asynccnt 0" ::: "memory");
  __syncthreads();

  const float total = scdf[MPART - 1];
  const size_t nbase = (size_t)n << 14;

#pragma unroll 1
  for (int i = 0; i < MPART / 256; ++i) {
    const int j = i * 256 + t;
    const size_t gid = nbase + j;
    const float target = u01((uint32_t)gid * 747796405u + 42u) * total;
    int lo = 0, hi = MPART - 1;
#pragma unroll 1
    while (lo < hi) {
      int mid = (lo + hi) >> 1;
      if (scdf[mid] < target) lo = mid + 1; else hi = mid;
    }
    const v4f s = *(const v4f*)(spred + (nbase + lo) * SDIM);
    __builtin_nontemporal_store(s, (v4f*)(out_states + gid * SDIM));
    __builtin_nontemporal_store(-9.7040605278f, out_lw + gid);
  }
}

// ---------------------------------------------------------------------------
extern "C" void kernel_launch(void* const* d_in, const int* in_sizes, int n_in,
                              void* d_out, int out_size, void* d_ws,
                              size_t ws_size, hipStream_t stream) {
  (void)in_sizes; (void)n_in; (void)out_size; (void)ws_size;
  const float* sprev = (const float*)d_in[0];
  const float* lwprev = (const float*)d_in[1];
  const float* obs = (const float*)d_in[2];
  const float* ctrl = (const float*)d_in[3];
  const float* noise = (const float*)d_in[4];
  const float* Adyn = (const float*)d_in[5];
  const float* Bm = (const float*)d_in[6];
  const float* Wm = (const float*)d_in[7];

  float* out = (float*)d_out;
  float* out_best = out;                                     // [512,4]
  float* out_states = out + (size_t)NROWS * SDIM;            // [512,16384,4]
  float* out_lw = out_states + (size_t)NROWS * MPART * SDIM; // [512,16384]

  // workspace layout (~168 MB)
  float* ws = (float*)d_ws;
  float* spred = ws;                                    // N*M*4
  float* lwws = spred + (size_t)NROWS * MPART * SDIM;   // N*M (-> CDF)
  float* rowc = lwws + (size_t)NROWS * MPART;           // N*8
  float* partials = rowc + NROWS * 8;                   // N*8*6
  float* lsews = partials + NROWS * BLOCKS_PER_ROW * 6; // N

  k_rowconst<<<2, 256, 0, stream>>>(obs, ctrl, Bm, Wm, rowc);
  k_main<<<NROWS * BLOCKS_PER_ROW, 256, 0, stream>>>(
      sprev, lwprev, noise, Adyn, rowc, spred, lwws, partials);
  k_finalize<<<2, 256, 0, stream>>>(partials, out_best, lsews);
  k_cdf<<<NROWS, 256, 0, stream>>>(lwws, lsews);
  k_resample<<<NROWS, 256, 0, stream>>>(spred, lwws, out_states, out_lw);
}